// SpatialModel_31997506355476
// MI455X (gfx1250) — compile-verified
//
#include <hip/hip_runtime.h>

// out[b,i,j,c] = sum_k prox[i,j,k] * x[b,sx[k],sy[k],c] * W[c]
// Per batch: O(225x7) = P(225x7) @ M_b(7x7), M_b[k,c] = x[b,sx[k],sy[k],c]*W[c]
// V_WMMA_F32_16X16X4_F32 (exact f32), two K-steps for K=7, TWO batches packed
// into the 16 N columns (cols 0-6 -> batch b, 7-13 -> batch b+1, 14-15 zero).

#define GRID_N   15
#define NSTA     7
#define ROWS     225            // GRID_N*GRID_N
#define BSTRIDE  (ROWS * NSTA)  // 1575 floats per batch
#define PAIRS    32             // 32 batch-pairs = 64 batches per wave

typedef __attribute__((ext_vector_type(2))) float v2f;
typedef __attribute__((ext_vector_type(8))) float v8f;

__global__ __launch_bounds__(32)
void spatial_wmma_f32(const float* __restrict__ x,
                      const float* __restrict__ W,
                      const float* __restrict__ prox,
                      const int*   __restrict__ sx,
                      const int*   __restrict__ sy,
                      float*       __restrict__ out)
{
    const int lane = threadIdx.x;          // wave32
    const int tile = blockIdx.x % 15;      // M-tile of P (rows 16*tile..16*tile+15)
    const int grp  = blockIdx.x / 15;      // group of 64 batches (all full groups)
    const int b0   = grp * (2 * PAIRS);

    const int h = lane >> 4;               // half-wave
    const int m = lane & 15;               // M (A rows) / N (B,D cols)

    // K index per (operand vgpr slot, half) for f32 16x16x4:
    //   slot0: h?2:0  slot1: h?3:1  |  slot2: h?6:4  slot3: h?7:5 (K=7 padded)
    const int kk[4] = { h ? 2 : 0, h ? 3 : 1, h ? 6 : 4, h ? 7 : 5 };

    // ---- A operand: rows of P = prox.reshape(225,7); loop-invariant ----
    const int r = tile * 16 + m;
    float a[4];
#pragma unroll
    for (int v = 0; v < 4; ++v) {
        const int k = kk[v];
        a[v] = (r < ROWS && k < NSTA) ? prox[r * NSTA + k] : 0.0f;
    }
    const v2f a01 = { a[0], a[1] };
    const v2f a23 = { a[2], a[3] };

    // ---- B column n = m encodes (batch offset db = m/7, channel cc = m%7) ----
    const bool colok = (m < 14);
    const int  db = colok ? (m / 7) : 0;
    const int  cc = colok ? (m % 7) : 0;

    // Loop-invariant per-lane gather offsets + per-slot weight (0 for padding).
    int   xoff[4];
    float mulv[4];
#pragma unroll
    for (int v = 0; v < 4; ++v) {
        const int  k   = kk[v];
        const bool val = colok && (k < NSTA);
        const int  si  = val ? sx[k] : 0;
        const int  sj  = val ? sy[k] : 0;
        xoff[v] = db * BSTRIDE + (si * GRID_N + sj) * NSTA + cc; // always in-bounds
        mulv[v] = val ? W[cc] : 0.0f;
    }

    // Loop-invariant per-lane store offsets (element units from pair base).
    int ooff[8];
#pragma unroll
    for (int v = 0; v < 8; ++v) {
        const int row = tile * 16 + v + 8 * h;
        ooff[v] = db * BSTRIDE + row * NSTA + cc;   // used only when valid
    }

    const float* xb = x   + (size_t)b0 * BSTRIDE;
    float*       ob = out + (size_t)b0 * BSTRIDE;

    // ---- software pipeline: preload pair 0 ----
    float raw[4];
#pragma unroll
    for (int v = 0; v < 4; ++v) raw[v] = xb[xoff[v]];

    if (tile < 14) {
        // All 16 rows valid: single exec toggle per iteration, 8 stores.
#pragma unroll
        for (int it = 0; it < PAIRS; ++it) {
            float bm[4];
#pragma unroll
            for (int v = 0; v < 4; ++v) bm[v] = raw[v] * mulv[v];
            const v2f bv01 = { bm[0], bm[1] };
            const v2f bv23 = { bm[2], bm[3] };

            if (it + 1 < PAIRS) {          // compile-time guard (full unroll)
                const float* xn = xb + (size_t)(it + 1) * (2 * BSTRIDE);
#pragma unroll
                for (int v = 0; v < 4; ++v) raw[v] = xn[xoff[v]];
            }

            v8f acc = {};
            acc = __builtin_amdgcn_wmma_f32_16x16x4_f32(
                      false, a01, false, bv01, (short)0, acc, false, false);
            acc = __builtin_amdgcn_wmma_f32_16x16x4_f32(
                      false, a23, false, bv23, (short)0, acc, false, false);

            float* op = ob + (size_t)it * (2 * BSTRIDE);
            if (colok) {
#pragma unroll
                for (int v = 0; v < 8; ++v) op[ooff[v]] = acc[v];
            }
        }
    } else {
        // tile 14: only row 224 valid (v=0, h=0).
#pragma unroll
        for (int it = 0; it < PAIRS; ++it) {
            float bm[4];
#pragma unroll
            for (int v = 0; v < 4; ++v) bm[v] = raw[v] * mulv[v];
            const v2f bv01 = { bm[0], bm[1] };
            const v2f bv23 = { bm[2], bm[3] };

            if (it + 1 < PAIRS) {
                const float* xn = xb + (size_t)(it + 1) * (2 * BSTRIDE);
#pragma unroll
                for (int v = 0; v < 4; ++v) raw[v] = xn[xoff[v]];
            }

            v8f acc = {};
            acc = __builtin_amdgcn_wmma_f32_16x16x4_f32(
                      false, a01, false, bv01, (short)0, acc, false, false);
            acc = __builtin_amdgcn_wmma_f32_16x16x4_f32(
                      false, a23, false, bv23, (short)0, acc, false, false);

            float* op = ob + (size_t)it * (2 * BSTRIDE);
            if (colok && h == 0) op[ooff[0]] = acc[0];
        }
    }
}

// Scalar fallback for a (normally empty) batch tail: one thread per (b,row),
// 7 channel outputs each. Only launched when nBatch % (2*PAIRS) != 0.
__global__ void spatial_tail(const float* __restrict__ x,
                             const float* __restrict__ W,
                             const float* __restrict__ prox,
                             const int*   __restrict__ sx,
                             const int*   __restrict__ sy,
                             float*       __restrict__ out,
                             int bStart, int nBatch)
{
    const int gid  = blockIdx.x * blockDim.x + threadIdx.x;
    const int nOut = (nBatch - bStart) * ROWS;
    if (gid >= nOut) return;
    const int b   = bStart + gid / ROWS;
    const int row = gid % ROWS;

    float s[NSTA];
#pragma unroll
    for (int k = 0; k < NSTA; ++k) {
        // s[k][c] gathered lazily below; cache station base offsets
        s[k] = 0.0f; // placeholder (overwritten per channel)
    }
    const float* xb = x + (size_t)b * BSTRIDE;
#pragma unroll
    for (int c = 0; c < NSTA; ++c) {
        float acc = 0.0f;
#pragma unroll
        for (int k = 0; k < NSTA; ++k) {
            const float xv = xb[(sx[k] * GRID_N + sy[k]) * NSTA + c];
            acc += prox[row * NSTA + k] * xv;
        }
        out[(size_t)b * BSTRIDE + row * NSTA + c] = acc * W[c];
    }
    (void)s;
}

extern "C" void kernel_launch(void* const* d_in, const int* in_sizes, int n_in,
                              void* d_out, int out_size, void* d_ws, size_t ws_size,
                              hipStream_t stream)
{
    const float* x    = (const float*)d_in[0];  // (B,15,15,7) f32
    const float* W    = (const float*)d_in[1];  // (7,) f32
    const float* prox = (const float*)d_in[2];  // (15,15,7) f32
    const int*   sx   = (const int*)d_in[3];    // (7,) int
    const int*   sy   = (const int*)d_in[4];    // (7,) int
    float*       out  = (float*)d_out;          // (B,15,15,7) f32

    const int nBatch = in_sizes[0] / BSTRIDE;   // 32768
    const int groups = nBatch / (2 * PAIRS);    // full 64-batch groups

    if (groups > 0) {
        dim3 grid(15 * groups);
        dim3 block(32);
        spatial_wmma_f32<<<grid, block, 0, stream>>>(x, W, prox, sx, sy, out);
    }

    const int bTail = groups * (2 * PAIRS);
    if (bTail < nBatch) {                       // not taken for B = 32768
        const int nOut = (nBatch - bTail) * ROWS;
        dim3 grid((nOut + 255) / 256);
        dim3 block(256);
        spatial_tail<<<grid, block, 0, stream>>>(x, W, prox, sx, sy, out,
                                                 bTail, nBatch);
    }
}